// SelectiveSSM_79139067396809
// MI455X (gfx1250) — compile-verified
//
#include <hip/hip_runtime.h>
#include <hip/hip_bf16.h>
#include <math.h>

#define B_    8
#define L_    1024
#define D_    512
#define N_    16
#define MTOT  (B_*L_)      // 8192 rows
#define C_    16           // scan chunks
#define CL_   64           // timesteps per chunk (C_*CL_ == L_)

typedef __attribute__((ext_vector_type(2))) float v2f;
typedef __attribute__((ext_vector_type(8))) float v8f;

__device__ __forceinline__ float silu_f(float v) { return v / (1.0f + __expf(-v)); }

// CDNA5 async global->LDS copy (4B per lane), tracked by ASYNCcnt.
__device__ __forceinline__ void async_g2l_b32(unsigned lds_off, const float* gaddr) {
  asm volatile("global_load_async_to_lds_b32 %0, %1, off"
               :: "v"(lds_off), "v"(gaddr) : "memory");
}
__device__ __forceinline__ void wait_async0() {
  asm volatile("s_wait_asynccnt 0x0" ::: "memory");
}

// ---------------------------------------------------------------------------
// Kernel 1: depthwise conv3 (SAME, zero pad) + SiLU  ->  u (B,L,D)
// ---------------------------------------------------------------------------
__global__ void k_conv_silu(const float* __restrict__ x,
                            const float* __restrict__ ck,   // (K=3,1,D)
                            float* __restrict__ u) {
  int idx = blockIdx.x * blockDim.x + threadIdx.x;   // over B*L*D
  if (idx >= B_ * L_ * D_) return;
  int d = idx % D_;
  int l = (idx / D_) % L_;
  float c0 = ck[0 * D_ + d], c1 = ck[1 * D_ + d], c2 = ck[2 * D_ + d];
  float xm = (l > 0)      ? x[idx - D_] : 0.0f;
  float xc = x[idx];
  float xp = (l < L_ - 1) ? x[idx + D_] : 0.0f;
  u[idx] = silu_f(c0 * xm + c1 * xc + c2 * xp);
}

// ---------------------------------------------------------------------------
// Kernel 2: proj = u @ [W_BC | W_dt] + [b_BC | b_dt]     (M=8192,K=512,N=64)
// One wave per 16x16 tile, V_WMMA_F32_16X16X4_F32, K-loop of 128.
// proj row layout: cols 0..15 = Bc, 16..31 = Cc, 32..63 = dt_rank proj.
// ---------------------------------------------------------------------------
__global__ void k_proj1(const float* __restrict__ u,
                        const float* __restrict__ W_BC, const float* __restrict__ b_BC,
                        const float* __restrict__ W_dt, const float* __restrict__ b_dt,
                        float* __restrict__ proj) {
  const int NT = 4;                                   // 64 / 16
  int wave = blockIdx.x * (blockDim.x >> 5) + (threadIdx.x >> 5);
  int lane = threadIdx.x & 31;
  if (wave >= (MTOT / 16) * NT) return;               // wave-uniform (never taken)
  int nt = wave % NT;
  int mt = wave / NT;
  int row0 = mt * 16;
  int ln = lane & 15;                                 // M for A-frag, N for B-frag
  int kk = (lane >> 4) << 1;                          // K sub-offset: 0 or 2

  const float* W = (nt < 2) ? W_BC : W_dt;            // both row-major (512,32)
  int ncol0 = (nt & 1) * 16;

  v8f acc = {};
  const float* arow = u + (size_t)(row0 + ln) * D_;
  for (int k = 0; k < D_; k += 4) {
    v2f a, b;
    a.x = arow[k + kk];
    a.y = arow[k + kk + 1];
    b.x = W[(k + kk)     * 32 + ncol0 + ln];
    b.y = W[(k + kk + 1) * 32 + ncol0 + ln];
    acc = __builtin_amdgcn_wmma_f32_16x16x4_f32(false, a, false, b,
                                                (short)0, acc, false, false);
  }
  int half = lane >> 4;
  int gcol = nt * 16 + ln;
  float bias = (gcol < 32) ? b_BC[gcol] : b_dt[gcol - 32];
#pragma unroll
  for (int r = 0; r < 8; ++r) {
    int row = row0 + r + 8 * half;
    proj[(size_t)row * 64 + gcol] = acc[r] + bias;
  }
}

// ---------------------------------------------------------------------------
// Kernel 3: Delta = softplus(dtr @ W_dtp + b_dtp)        (M=8192,K=32,N=512)
// ---------------------------------------------------------------------------
__global__ void k_proj2(const float* __restrict__ proj,
                        const float* __restrict__ W_dtp,   // (32,512) row-major
                        const float* __restrict__ b_dtp,
                        float* __restrict__ Delta) {
  const int NT = 32;                                  // 512 / 16
  int wave = blockIdx.x * (blockDim.x >> 5) + (threadIdx.x >> 5);
  int lane = threadIdx.x & 31;
  if (wave >= (MTOT / 16) * NT) return;               // wave-uniform (never taken)
  int nt = wave % NT;
  int mt = wave / NT;
  int row0 = mt * 16;
  int n0 = nt * 16;
  int ln = lane & 15;
  int kk = (lane >> 4) << 1;

  v8f acc = {};
  const float* arow = proj + (size_t)(row0 + ln) * 64 + 32;
#pragma unroll
  for (int k = 0; k < 32; k += 4) {
    v2f a, b;
    a.x = arow[k + kk];
    a.y = arow[k + kk + 1];
    b.x = W_dtp[(k + kk)     * D_ + n0 + ln];
    b.y = W_dtp[(k + kk + 1) * D_ + n0 + ln];
    acc = __builtin_amdgcn_wmma_f32_16x16x4_f32(false, a, false, b,
                                                (short)0, acc, false, false);
  }
  int half = lane >> 4;
  float bias = b_dtp[n0 + ln];
#pragma unroll
  for (int r = 0; r < 8; ++r) {
    int row = row0 + r + 8 * half;
    float v = acc[r] + bias;
    float sp = (v > 20.0f) ? v : log1pf(__expf(v));   // stable softplus
    Delta[(size_t)row * D_ + n0 + ln] = sp;
  }
}

// ---------------------------------------------------------------------------
// Chunked scan. Recurrence h[l] = a[l]*h[l-1] + b[l] composes per chunk as
//   h_out = P*h_in + g     (P = prod a, g = local scan with h_in = 0)
// Pass A: per (b,d,chunk) compute P[n], g[n].           65536 thr = 2048 waves
// Pass B: per (b,d) compose 16 carries -> Hin[chunk].   4096 thr
// Pass C: rerun chunk from Hin, emit y + x*Dparam.      65536 thr = 2048 waves
// P/G/Hin layout: [b][c][n][d]  (d fastest -> coalesced in every pass)
// ---------------------------------------------------------------------------
#define PGIDX(b,c,n,d) ((((size_t)(b)*C_ + (c))*N_ + (n))*D_ + (d))
#define LOG2E_ 1.4426950408889634f

__global__ void k_scan_partial(const float* __restrict__ x,
                               const float* __restrict__ proj,
                               const float* __restrict__ Delta,
                               const float* __restrict__ A_log,
                               float* __restrict__ Pbuf, float* __restrict__ Gbuf) {
  int blk = blockIdx.x;                       // B_*C_*(D_/256)
  int dg = blk % (D_ / 256);
  int c  = (blk / (D_ / 256)) % C_;
  int b  = blk / ((D_ / 256) * C_);
  int d  = dg * 256 + threadIdx.x;

  __shared__ float sB[CL_ * 16];              // Bc cols only: 4 KB
  const float* bcbase = proj + (size_t)(b * L_ + c * CL_) * 64;
  for (int t = threadIdx.x; t < CL_ * 16; t += 256) {
    int lr = t >> 4, cc = t & 15;
    async_g2l_b32((unsigned)(t * 4) + (unsigned)(size_t)sB,
                  bcbase + (size_t)lr * 64 + cc);
  }
  float A2[N_];
#pragma unroll
  for (int n = 0; n < N_; ++n)
    A2[n] = -__expf(A_log[d * N_ + n]) * LOG2E_;
  wait_async0();
  __syncthreads();

  float h[N_], p[N_];
#pragma unroll
  for (int n = 0; n < N_; ++n) { h[n] = 0.0f; p[n] = 1.0f; }

  for (int li = 0; li < CL_; ++li) {
    size_t gidx = (size_t)(b * L_ + c * CL_ + li) * D_ + d;
    float dv = Delta[gidx];
    float xd = x[gidx] * dv;
    const float* brow = &sB[li * 16];
#pragma unroll
    for (int n = 0; n < N_; ++n) {
      float alpha = exp2f(A2[n] * dv);
      p[n] *= alpha;
      h[n] = alpha * h[n] + brow[n] * xd;
    }
  }
#pragma unroll
  for (int n = 0; n < N_; ++n) {
    Pbuf[PGIDX(b, c, n, d)] = p[n];
    Gbuf[PGIDX(b, c, n, d)] = h[n];
  }
}

__global__ void k_scan_carry(const float* __restrict__ Pbuf,
                             const float* __restrict__ Gbuf,
                             float* __restrict__ Hin) {
  int b = blockIdx.x / (D_ / 256);
  int d = (blockIdx.x % (D_ / 256)) * 256 + threadIdx.x;
  float h[N_];
#pragma unroll
  for (int n = 0; n < N_; ++n) h[n] = 0.0f;
  for (int c = 0; c < C_; ++c) {
#pragma unroll
    for (int n = 0; n < N_; ++n) {
      size_t i = PGIDX(b, c, n, d);
      Hin[i] = h[n];                          // state entering chunk c
      h[n] = Pbuf[i] * h[n] + Gbuf[i];        // compose carry
    }
  }
}

__global__ void k_scan_final(const float* __restrict__ x,
                             const float* __restrict__ proj,
                             const float* __restrict__ Delta,
                             const float* __restrict__ A_log,
                             const float* __restrict__ Dparam,
                             const float* __restrict__ Hin,
                             float* __restrict__ out) {
  int blk = blockIdx.x;
  int dg = blk % (D_ / 256);
  int c  = (blk / (D_ / 256)) % C_;
  int b  = blk / ((D_ / 256) * C_);
  int d  = dg * 256 + threadIdx.x;

  __shared__ float sBC[CL_ * 32];             // Bc|Cc: 8 KB
  const float* bcbase = proj + (size_t)(b * L_ + c * CL_) * 64;
  for (int t = threadIdx.x; t < CL_ * 32; t += 256) {
    int lr = t >> 5, cc = t & 31;
    async_g2l_b32((unsigned)(t * 4) + (unsigned)(size_t)sBC,
                  bcbase + (size_t)lr * 64 + cc);
  }
  float A2[N_];
#pragma unroll
  for (int n = 0; n < N_; ++n)
    A2[n] = -__expf(A_log[d * N_ + n]) * LOG2E_;
  float Dp = Dparam[d];
  float h[N_];
#pragma unroll
  for (int n = 0; n < N_; ++n) h[n] = Hin[PGIDX(b, c, n, d)];
  wait_async0();
  __syncthreads();

  for (int li = 0; li < CL_; ++li) {
    size_t gidx = (size_t)(b * L_ + c * CL_ + li) * D_ + d;
    float dv = Delta[gidx];
    float xv = x[gidx];
    float xd = xv * dv;
    const float* bc = &sBC[li * 32];
    float y = 0.0f;
#pragma unroll
    for (int n = 0; n < N_; ++n) {
      float alpha = exp2f(A2[n] * dv);
      h[n] = alpha * h[n] + bc[n] * xd;
      y   += bc[16 + n] * h[n];
    }
    out[gidx] = y + xv * Dp;
  }
}

// ---------------------------------------------------------------------------
extern "C" void kernel_launch(void* const* d_in, const int* in_sizes, int n_in,
                              void* d_out, int out_size, void* d_ws, size_t ws_size,
                              hipStream_t stream) {
  const float* x    = (const float*)d_in[0];
  const float* ck   = (const float*)d_in[1];
  const float* W_BC = (const float*)d_in[2];
  const float* b_BC = (const float*)d_in[3];
  const float* A_lg = (const float*)d_in[4];
  const float* Dpar = (const float*)d_in[5];
  const float* W_dt = (const float*)d_in[6];
  const float* b_dt = (const float*)d_in[7];
  const float* Wdtp = (const float*)d_in[8];
  const float* bdtp = (const float*)d_in[9];
  float* out = (float*)d_out;

  char* ws = (char*)d_ws;
  const size_t SZ_U  = (size_t)B_ * L_ * D_ * 4;          // 16 MB
  const size_t SZ_PG = (size_t)B_ * C_ * N_ * D_ * 4;     //  4 MB
  float* u     = (float*)ws;                              // [0,16MB) (dead after proj1)
  float* Pbuf  = (float*)ws;                              // aliases u: [0,4MB)
  float* Gbuf  = (float*)(ws + SZ_PG);                    // aliases u: [4,8MB)
  float* proj  = (float*)(ws + SZ_U);                     //  2 MB
  float* Delta = (float*)(ws + SZ_U + (size_t)MTOT * 64 * 4);   // 16 MB
  float* Hin   = (float*)(ws + SZ_U + (size_t)MTOT * 64 * 4 + SZ_U); // 4 MB

  // 1) conv + silu
  k_conv_silu<<<(B_ * L_ * D_ + 255) / 256, 256, 0, stream>>>(x, ck, u);

  // 2) proj = u @ [W_BC|W_dt] + bias : 2048 waves
  k_proj1<<<512, 128, 0, stream>>>(u, W_BC, b_BC, W_dt, b_dt, proj);

  // 3) Delta = softplus(dtr @ W_dtp + b_dtp) : 16384 waves
  k_proj2<<<4096, 128, 0, stream>>>(proj, Wdtp, bdtp, Delta);

  // 4) chunked scan: partial -> carry -> final
  int scan_blocks = B_ * C_ * (D_ / 256);                 // 256 blocks
  k_scan_partial<<<scan_blocks, 256, 0, stream>>>(x, proj, Delta, A_lg, Pbuf, Gbuf);
  k_scan_carry<<<B_ * (D_ / 256), 256, 0, stream>>>(Pbuf, Gbuf, Hin);
  k_scan_final<<<scan_blocks, 256, 0, stream>>>(x, proj, Delta, A_lg, Dpar, Hin, out);
}